// GCNEncoder_84507776516794
// MI455X (gfx1250) — compile-verified
//
#include <hip/hip_runtime.h>

typedef float v2f __attribute__((ext_vector_type(2)));
typedef float v8f __attribute__((ext_vector_type(8)));

// ---------------------------------------------------------------- utilities

__global__ void fill_f32(float* __restrict__ p, float v, int n) {
  int i = blockIdx.x * blockDim.x + threadIdx.x;
  if (i < n) p[i] = v;
}

__global__ void count_deg(const int* __restrict__ ei, float* __restrict__ deg, int E) {
  int e = blockIdx.x * blockDim.x + threadIdx.x;
  if (e < E) atomicAdd(&deg[ei[E + e]], 1.0f);   // dst = ei[1][e]
}

__global__ void finalize_dinv(float* __restrict__ deg, int n) {
  int i = blockIdx.x * blockDim.x + threadIdx.x;
  if (i < n) deg[i] = rsqrtf(deg[i]);
}

// ------------------------------------------------- fp32 WMMA GEMM (16x16x4)
// H[M,N] = X[M,K] @ W[K,N], row-major. Requires M%16==0, K%4==0, N%16==0.
// One wave per 16x16 output tile; EXEC is uniformly all-1s (grid sized exactly).
__global__ void gemm_wmma_f32(const float* __restrict__ X, const float* __restrict__ W,
                              float* __restrict__ H, int M, int K, int N) {
  int wavesPerBlock = blockDim.x >> 5;
  int wave = blockIdx.x * wavesPerBlock + (threadIdx.x >> 5);
  int lane = threadIdx.x & 31;
  int tilesN = N >> 4;
  int tm = wave / tilesN;
  int tn = wave - tm * tilesN;
  if (tm * 16 >= M) return;                 // wave-uniform guard

  int r = lane & 15;                        // A row / B,D column within tile
  int h = lane >> 4;                        // half-wave selector

  const float* xrow = X + (size_t)(tm * 16 + r) * K + 2 * h;       // A: a[v]=X[m][k0+2h+v]
  const float* wcol = W + (size_t)(2 * h) * N + tn * 16 + r;       // B: b[v]=W[k0+2h+v][n]

  v8f acc = {};
  for (int k0 = 0; k0 < K; k0 += 4) {
    v2f a, b;
    a.x = xrow[k0];
    a.y = xrow[k0 + 1];
    b.x = wcol[(size_t)k0 * N];
    b.y = wcol[(size_t)(k0 + 1) * N];
    // 8 args: (neg_a, A, neg_b, B, c_mod, C, reuse_a, reuse_b)
    acc = __builtin_amdgcn_wmma_f32_16x16x4_f32(false, a, false, b,
                                                (short)0, acc, false, false);
  }

  // D layout: d[v] -> row (8h + v), col r
  float* drow = H + (size_t)(tm * 16 + 8 * h) * N + tn * 16 + r;
#pragma unroll
  for (int v = 0; v < 8; ++v) drow[(size_t)v * N] = acc[v];
}

// --------------------------------------------------- edge scatter (gather + atomicAdd)
// One thread per (edge, 4-feature chunk); consecutive threads hit consecutive
// features of the same edge -> coalesced float4 gather, coalesced atomics.
__global__ void scatter_edges(const int* __restrict__ ei, const float* __restrict__ h,
                              const float* __restrict__ dinv, float* __restrict__ agg,
                              int E, int logF4, int F) {
  int tid = blockIdx.x * blockDim.x + threadIdx.x;
  if (tid >= (E << logF4)) return;
  int e  = tid >> logF4;
  int f4 = tid & ((1 << logF4) - 1);
  int s = ei[e];         // src
  int d = ei[E + e];     // dst
  float c = dinv[s] * dinv[d];
  float4 hv = ((const float4*)(h + (size_t)s * F))[f4];
  float* ap = agg + (size_t)d * F + (f4 << 2);
  atomicAdd(ap + 0, hv.x * c);
  atomicAdd(ap + 1, hv.y * c);
  atomicAdd(ap + 2, hv.z * c);
  atomicAdd(ap + 3, hv.w * c);
}

// ------------------------------------- combine: agg + h*dinv^2 + bias (+ReLU)
__global__ void combine_nodes(const float* __restrict__ agg, const float* __restrict__ hm,
                              const float* __restrict__ dinv, const float* __restrict__ bias,
                              float* __restrict__ out, int total, int logF, int relu) {
  int tid = blockIdx.x * blockDim.x + threadIdx.x;
  if (tid >= total) return;
  int i = tid >> logF;
  int f = tid & ((1 << logF) - 1);
  float di = dinv[i];
  float v = agg[tid] + hm[tid] * di * di + bias[f];
  out[tid] = relu ? fmaxf(v, 0.0f) : v;
}

// ---------------------------------------------------------------- launcher

extern "C" void kernel_launch(void* const* d_in, const int* in_sizes, int n_in,
                              void* d_out, int out_size, void* d_ws, size_t ws_size,
                              hipStream_t stream) {
  const float* x  = (const float*)d_in[0];
  const int*   ei = (const int*)d_in[1];
  const float* W1 = (const float*)d_in[2];
  const float* b1 = (const float*)d_in[3];
  const float* W2 = (const float*)d_in[4];
  const float* b2 = (const float*)d_in[5];

  const int DIN = 64, DHID = 128, DOUT = 64;
  const int N = in_sizes[0] / DIN;        // 100000 (divisible by 16)
  const int E = in_sizes[1] / 2;          // 1600000

  float* dinv = (float*)d_ws;             // N floats
  float* bufA = dinv + N;                 // N*DHID floats (h1, then h2)
  float* bufB = bufA + (size_t)N * DHID;  // N*DHID floats (agg1 -> relu'd out1)
  float* out  = (float*)d_out;            // N*DOUT floats

  const int T = 256;

  // degrees: deg = 1 (self-loop) + in-degree; then dinv = rsqrt(deg)
  fill_f32<<<(N + T - 1) / T, T, 0, stream>>>(dinv, 1.0f, N);
  count_deg<<<(E + T - 1) / T, T, 0, stream>>>(ei, dinv, E);
  finalize_dinv<<<(N + T - 1) / T, T, 0, stream>>>(dinv, N);

  // ---- layer 1: h1 = x @ W1  (WMMA f32)
  {
    int waves = (N / 16) * (DHID / 16);   // 50000, divisible by 8
    gemm_wmma_f32<<<waves / 8, T, 0, stream>>>(x, W1, bufA, N, DIN, DHID);
  }
  fill_f32<<<((N * DHID) + T - 1) / T, T, 0, stream>>>(bufB, 0.0f, N * DHID);
  scatter_edges<<<((E * (DHID / 4)) + T - 1) / T, T, 0, stream>>>(
      ei, bufA, dinv, bufB, E, 5 /*log2(128/4)*/, DHID);
  combine_nodes<<<((N * DHID) + T - 1) / T, T, 0, stream>>>(
      bufB, bufA, dinv, b1, bufB, N * DHID, 7 /*log2(128)*/, 1);

  // ---- layer 2: h2 = relu_out @ W2  (WMMA f32), h2 overwrites bufA
  {
    int waves = (N / 16) * (DOUT / 16);   // 25000, divisible by 8
    gemm_wmma_f32<<<waves / 8, T, 0, stream>>>(bufB, W2, bufA, N, DHID, DOUT);
  }
  fill_f32<<<((N * DOUT) + T - 1) / T, T, 0, stream>>>(out, 0.0f, N * DOUT);
  scatter_edges<<<((E * (DOUT / 4)) + T - 1) / T, T, 0, stream>>>(
      ei, bufA, dinv, out, E, 4 /*log2(64/4)*/, DOUT);
  combine_nodes<<<((N * DOUT) + T - 1) / T, T, 0, stream>>>(
      out, bufA, dinv, b2, out, N * DOUT, 6 /*log2(64)*/, 0);
}